// MCnet_77438260347512
// MI455X (gfx1250) — compile-verified
//
#include <hip/hip_runtime.h>

// Problem constants (match reference setup_inputs).
#define BB 16
#define HH 256
#define WW 256
#define SS 4

typedef float v4f __attribute__((ext_vector_type(4)));

// ---------------------------------------------------------------------------
// Draft kernel: one block per (output row y, batch b, frame t).
// Stages flow[b, 0:2, 4y:4y+4, :] (32KB) into LDS via CDNA5 async loads
// (non-temporal: the flow is streamed exactly once), then computes all 16
// phases for that row (one output column per thread). Image gathers stay
// temporal so x (12.6MB) lives in WGP$/L2; draft stores are non-temporal.
// ---------------------------------------------------------------------------
__global__ __launch_bounds__(256) void mc_drafts_kernel(
    const float* __restrict__ xin,     // [16,3,256,256]
    const float* __restrict__ flow01,  // [16,2,1024,1024]
    const float* __restrict__ flow21,  // [16,2,1024,1024]
    float* __restrict__ out)           // [16,35,256,256]
{
    // [comp][row 0..3][col 0..1023] = 8192 floats = 32 KB
    __shared__ float sflow[2 * 4 * 1024];

    const int y   = blockIdx.x;   // output row
    const int b   = blockIdx.y;   // batch
    const int t   = blockIdx.z;   // 0 -> warp x[:,0] with flow01 ; 1 -> x[:,2] with flow21
    const int tid = threadIdx.x;  // output column

    const float* __restrict__ flowsrc = (t == 0) ? flow01 : flow21;
    const int y4 = y * 4;

    // ---- Async stage of the flow tile into LDS (coalesced b128 per lane) ----
    // 8192 floats -> 2048 16-byte chunks -> 8 chunks per thread.
#pragma unroll
    for (int it = 0; it < 8; ++it) {
        const int q    = it * 256 + tid;   // chunk id
        const int f    = q * 4;            // float index within tile
        const int comp = f >> 12;          // /4096
        const int rem  = f & 4095;
        const int r    = rem >> 10;        // tile row 0..3
        const int X    = rem & 1023;       // column
        const float* gp = flowsrc
            + ((((size_t)(b * 2 + comp)) * 1024 + (size_t)(y4 + r)) << 10) + X;
        const unsigned long long ga = (unsigned long long)gp;
        const unsigned lds = (unsigned)(unsigned long long)(&sflow[f]);
        asm volatile("global_load_async_to_lds_b128 %0, %1, off th:TH_LOAD_NT"
                     :: "v"(lds), "v"(ga)
                     : "memory");
    }
    asm volatile("s_wait_asynccnt 0" ::: "memory");
    __syncthreads();

    // ---- Compute: bilinear warp for all 16 phases, this row/column ----
    const int c = t * 2;  // source channel: 0 or 2
    const float* __restrict__ img = xin + (((size_t)b * 3 + c) << 16);  // 256*256

    const float bx = -1.0f + (2.0f / 255.0f) * (float)tid;  // linspace(-1,1,256)[x]
    const float by = -1.0f + (2.0f / 255.0f) * (float)y;
    const float k  = 31.0f / 255.0f;

    // out channel for phase p is 3 + 2p + t
    const size_t obase = ((((size_t)b * 35 + 3 + t) << 8) + (size_t)y) << 8;
    const int xc4 = tid * 4;

#pragma unroll
    for (int p = 0; p < 16; ++p) {
        const int i = p >> 2;   // row phase
        const int j = p & 3;    // col phase

        const float f0 = sflow[i * 1024 + xc4 + j] * 0.25f;           // comp 0
        const float f1 = sflow[4096 + i * 1024 + xc4 + j] * 0.25f;    // comp 1

        const float gx = bx + f0 * k;
        const float gy = by + f1 * k;
        float fx = (gx + 1.0f) * 127.5f;
        float fy = (gy + 1.0f) * 127.5f;
        fx = fminf(fmaxf(fx, 0.0f), 255.0f);
        fy = fminf(fmaxf(fy, 0.0f), 255.0f);

        const float x0f = floorf(fx);
        const float y0f = floorf(fy);
        const float wx = fx - x0f;
        const float wy = fy - y0f;
        const int x0 = (int)x0f;
        const int y0 = (int)y0f;
        const int x1 = min(x0 + 1, 255);
        const int y1 = min(y0 + 1, 255);

        const float v00 = img[(y0 << 8) + x0];
        const float v01 = img[(y0 << 8) + x1];
        const float v10 = img[(y1 << 8) + x0];
        const float v11 = img[(y1 << 8) + x1];

        const float top = v00 * (1.0f - wx) + v01 * wx;
        const float bot = v10 * (1.0f - wx) + v11 * wx;
        const float res = top * (1.0f - wy) + bot * wy;
        __builtin_nontemporal_store(res, &out[obase + ((size_t)(2 * p) << 16) + tid]);
    }
}

// ---------------------------------------------------------------------------
// Copy x into output channels 0..2 (vectorized). Loads stay temporal (the
// draft kernel re-reads x); stores are non-temporal (written once).
// ---------------------------------------------------------------------------
__global__ __launch_bounds__(256) void copy_x_kernel(
    const v4f* __restrict__ xin, v4f* __restrict__ outp)
{
    const int q = blockIdx.x * 256 + threadIdx.x;  // 0 .. 16*3*16384-1
    const int b = q / (3 * 16384);
    const int r = q - b * (3 * 16384);
    __builtin_nontemporal_store(xin[q], &outp[(size_t)b * (35 * 16384) + r]);
}

extern "C" void kernel_launch(void* const* d_in, const int* in_sizes, int n_in,
                              void* d_out, int out_size, void* d_ws, size_t ws_size,
                              hipStream_t stream) {
    (void)in_sizes; (void)n_in; (void)out_size; (void)d_ws; (void)ws_size;

    const float* x      = (const float*)d_in[0];
    const float* flow01 = (const float*)d_in[1];
    const float* flow21 = (const float*)d_in[2];
    float* out          = (float*)d_out;

    // Channels 0..2: passthrough copy of x. 16*3*65536/4 = 786432 float4.
    copy_x_kernel<<<786432 / 256, 256, 0, stream>>>(
        (const v4f*)x, (v4f*)out);

    // Channels 3..34: motion-compensated drafts.
    dim3 grid(HH, BB, 2);
    mc_drafts_kernel<<<grid, 256, 0, stream>>>(x, flow01, flow21, out);
}